// DWT2D_25580825215625
// MI455X (gfx1250) — compile-verified
//
#include <hip/hip_runtime.h>

// -----------------------------------------------------------------------------
// Fused 2D db4 DWT (pywt dwt2, 'symmetric' mode) for x:(8,64,512,512) fp32.
// out = [LL | LH | HL | HH], each (512 images, 259, 259) fp32.
//
// Row pass: scalar FMA into LDS. Column pass: banded matmul on the CDNA5
// matrix pipe via V_WMMA_F32_16X16X4_F32 (Z16x16 = F16x40 * Y40x16, 10 K=4
// chunks). Keeps everything fp32; kernel is HBM-bound (~1.4 GB traffic).
// -----------------------------------------------------------------------------

typedef __attribute__((ext_vector_type(2))) float v2f;
typedef __attribute__((ext_vector_type(8))) float v8f;

#define IH 512
#define IW 512
#define OH 259
#define OW 259
#define NIMG 512  // 8*64

// Reversed db4 decomposition filters: FR*[t] = DEC_*[7-t]
// out[i] = sum_t FR[t] * x_sym(2i + t - 6)
__device__ constexpr float FRLO[8] = {
     0.23037781330885523f,  0.7148465705525415f,   0.6308807679295904f,
    -0.02798376941698385f, -0.18703481171888114f,  0.030841381835986965f,
     0.032883011666982945f, -0.010597401784997278f };
__device__ constexpr float FRHI[8] = {
    -0.010597401784997278f, -0.032883011666982945f, 0.030841381835986965f,
     0.18703481171888114f,  -0.02798376941698385f, -0.6308807679295904f,
     0.7148465705525415f,   -0.23037781330885523f };

__device__ __forceinline__ int refl(int s, int n) {
    // numpy/pywt 'symmetric' reflection (edge repeated); overshoot < n here.
    if (s < 0)  s = -1 - s;
    if (s >= n) s = 2 * n - 1 - s;
    return s;
}

// Select filter tap d from compile-time table (0 outside [0,8)); folds to
// a v_cndmask chain, no memory traffic.
__device__ __forceinline__ float pick_lo(int d) {
    float r = 0.f;
#pragma unroll
    for (int j = 0; j < 8; ++j) r = (d == j) ? FRLO[j] : r;
    return r;
}
__device__ __forceinline__ float pick_hi(int d) {
    float r = 0.f;
#pragma unroll
    for (int j = 0; j < 8; ++j) r = (d == j) ? FRHI[j] : r;
    return r;
}

__global__ __launch_bounds__(128)
void dwt2_db4_kernel(const float* __restrict__ x, float* __restrict__ out)
{
    // LDS: input patch 38x134 (stride 136), row-pass result 2ch x 40 x 64
    // (stride 68 -> lanes 0-15 and 16-31 of a B-load hit disjoint banks).
    __shared__ float xs[38][136];
    __shared__ float ys[2][40][68];

    const int tid  = threadIdx.x;
    const int lane = tid & 31;           // wave32
    const int wave = tid >> 5;           // 4 waves, one 16-col strip each

    const int n0  = blockIdx.x * 64;     // output col tile base
    const int m0  = blockIdx.y * 16;     // output row tile base
    const int img = blockIdx.z;

    const float* xin = x + (size_t)img * (IH * IW);

    // ---- Phase 1: load symmetric-extended 38x134 patch ----------------------
    const int rbase = 2 * m0 - 6;
    const int cbase = 2 * n0 - 6;
    for (int idx = tid; idx < 38 * 134; idx += 128) {
        int rr = idx / 134, cc = idx - rr * 134;
        int gr = refl(rbase + rr, IH);
        int gc = refl(cbase + cc, IW);
        xs[rr][cc] = xin[gr * IW + gc];
    }
    __syncthreads();

    // ---- Phase 2: row pass (lo/hi), 38 rows x 64 cols; rows 38-39 zero -----
    for (int idx = tid; idx < 38 * 64; idx += 128) {
        int rr = idx >> 6, i = idx & 63;
        float slo = 0.f, shi = 0.f;
#pragma unroll
        for (int t = 0; t < 8; ++t) {
            float v = xs[rr][2 * i + t];
            slo = fmaf(FRLO[t], v, slo);
            shi = fmaf(FRHI[t], v, shi);
        }
        ys[0][rr][i] = slo;
        ys[1][rr][i] = shi;
    }
    for (int idx = tid; idx < 2 * 2 * 68; idx += 128) {   // zero K-pad rows
        int ch = idx / 136, rem = idx - ch * 136;
        ys[ch][38 + rem / 68][rem % 68] = 0.f;
    }
    __syncthreads();

    // ---- Phase 3: column pass as banded matmul on WMMA ---------------------
    // Z[m][n] = sum_k F[m][k] * Y[2m0-6 + k][n], F[m][k] = FR[k-2m] in-band.
    // A (16x4 f32) ISA layout: lanes 0-15 -> K={0,1} in VGPR0/1,
    // lanes 16-31 -> K={2,3}.  B assumed to mirror (K split across halves).
    const int mrow  = lane & 15;
    const int khalf = (lane >> 4) << 1;          // 0 or 2
    v2f alo[10], ahi[10];
#pragma unroll
    for (int c = 0; c < 10; ++c) {
        int d0 = 4 * c + khalf - 2 * mrow;
        alo[c].x = pick_lo(d0);     alo[c].y = pick_lo(d0 + 1);
        ahi[c].x = pick_hi(d0);     ahi[c].y = pick_hi(d0 + 1);
    }

    const int ncol = (wave << 4) + mrow;         // local output column
    v8f cLL = {}, cLH = {}, cHL = {}, cHH = {};
#pragma unroll
    for (int c = 0; c < 10; ++c) {
        int k0 = 4 * c + khalf;
        v2f blo, bhi;
        blo.x = ys[0][k0][ncol];  blo.y = ys[0][k0 + 1][ncol];
        bhi.x = ys[1][k0][ncol];  bhi.y = ys[1][k0 + 1][ncol];
        cLL = __builtin_amdgcn_wmma_f32_16x16x4_f32(false, alo[c], false, blo,
                                                    (short)0, cLL, false, false);
        cLH = __builtin_amdgcn_wmma_f32_16x16x4_f32(false, ahi[c], false, blo,
                                                    (short)0, cLH, false, false);
        cHL = __builtin_amdgcn_wmma_f32_16x16x4_f32(false, alo[c], false, bhi,
                                                    (short)0, cHL, false, false);
        cHH = __builtin_amdgcn_wmma_f32_16x16x4_f32(false, ahi[c], false, bhi,
                                                    (short)0, cHH, false, false);
    }

    // ---- Phase 4: store D tiles (lane0-15: M=v, lane16-31: M=v+8) ----------
    const size_t sub  = (size_t)NIMG * OH * OW;  // per-subband element stride
    float* base = out + (size_t)img * (OH * OW);
    const int colg = n0 + ncol;
    const int mhi  = (lane >> 4) << 3;
    if (colg < OW) {
#pragma unroll
        for (int v = 0; v < 8; ++v) {
            int rowg = m0 + v + mhi;
            if (rowg < OH) {
                size_t o = (size_t)rowg * OW + colg;
                base[o]            = cLL[v];   // LL
                base[o + sub]      = cLH[v];   // LH (cH)
                base[o + 2 * sub]  = cHL[v];   // HL (cV)
                base[o + 3 * sub]  = cHH[v];   // HH (cD)
            }
        }
    }
}

extern "C" void kernel_launch(void* const* d_in, const int* in_sizes, int n_in,
                              void* d_out, int out_size, void* d_ws, size_t ws_size,
                              hipStream_t stream) {
    const float* x = (const float*)d_in[0];
    float* out = (float*)d_out;
    dim3 grid((OW + 63) / 64, (OH + 15) / 16, NIMG);   // 5 x 17 x 512
    dwt2_db4_kernel<<<grid, dim3(128), 0, stream>>>(x, out);
}